// QuanvolutionClassifier_65481071407285
// MI455X (gfx1250) — compile-verified
//
#include <hip/hip_runtime.h>
#include <math.h>

typedef float v2f __attribute__((ext_vector_type(2)));
typedef float v8f __attribute__((ext_vector_type(8)));

#define N_PATCH 196
#define N_TILES 13  // ceil(196/16)

// ---------------------------------------------------------------------------
// Setup: from vparams build A_w[j][k] (quantum circuit collapsed to four 16x16
// quadratic forms) and emit it as a lane-packed WMMA B-operand table:
//   BPACK[lane*32 + ks*8 + t4*2 + v] = A_cat[4*ks + 2*(lane>>4) + v][16*t4 + (lane&15)]
// ---------------------------------------------------------------------------
__global__ __launch_bounds__(64) void quanv_setup_kernel(
    const float* __restrict__ vp, float* __restrict__ bpack) {
  __shared__ float Ur[16][16];  // [m][column j]
  __shared__ float Ui[16][16];
  __shared__ float Acat[1024];  // [j][w*16+k]
  const int t = threadIdx.x;
  if (t < 16) {
    float sr[16], si[16];
#pragma unroll
    for (int m = 0; m < 16; ++m) { sr[m] = 0.f; si[m] = 0.f; }
    sr[t] = 1.f;  // column t = circuit applied to |t>
#pragma unroll
    for (int l = 0; l < 3; ++l) {
#pragma unroll
      for (int i = 0; i < 4; ++i) {
        const int bit = 3 - i;  // wire 0 = MSB
        float sy, cy; sincosf(0.5f * vp[l * 8 + i], &sy, &cy);  // RY(p[i])
#pragma unroll
        for (int m = 0; m < 16; ++m) {
          if (((m >> bit) & 1) == 0) {
            const int m1 = m | (1 << bit);
            const float a0r = sr[m], a0i = si[m], a1r = sr[m1], a1i = si[m1];
            sr[m]  = cy * a0r - sy * a1r;  si[m]  = cy * a0i - sy * a1i;
            sr[m1] = sy * a0r + cy * a1r;  si[m1] = sy * a0i + cy * a1i;
          }
        }
        float sz, cz; sincosf(0.5f * vp[l * 8 + i + 1], &sz, &cz);  // RZ(p[i+1])
#pragma unroll
        for (int m = 0; m < 16; ++m) {
          const float ar = sr[m], ai = si[m];
          if (((m >> bit) & 1) == 0) { sr[m] = cz * ar + sz * ai; si[m] = cz * ai - sz * ar; }
          else                       { sr[m] = cz * ar - sz * ai; si[m] = cz * ai + sz * ar; }
        }
      }
#pragma unroll
      for (int i = 0; i < 4; ++i) {  // CX(i,(i+1)%4)
        const int cbit = 3 - i;
        const int tbit = 3 - ((i + 1) & 3);
#pragma unroll
        for (int m = 0; m < 16; ++m) {
          if (((m >> cbit) & 1) == 1 && ((m >> tbit) & 1) == 0) {
            const int m1 = m | (1 << tbit);
            float tr = sr[m]; sr[m] = sr[m1]; sr[m1] = tr;
            float ti = si[m]; si[m] = si[m1]; si[m1] = ti;
          }
        }
      }
    }
#pragma unroll
    for (int m = 0; m < 16; ++m) { Ur[m][t] = sr[m]; Ui[m][t] = si[m]; }
  }
  __syncthreads();
  for (int idx = t; idx < 1024; idx += 64) {  // A_w[j][k]
    const int j = idx >> 6;
    const int w = (idx >> 4) & 3;
    const int k = idx & 15;
    const int bit = 3 - w;
    float acc = 0.f;
#pragma unroll
    for (int m = 0; m < 16; ++m) {
      const float sgn = ((m >> bit) & 1) ? -1.f : 1.f;
      acc += sgn * (Ur[m][j] * Ur[m][k] + Ui[m][j] * Ui[m][k]);
    }
    Acat[idx] = acc;
  }
  __syncthreads();
  for (int i = t; i < 1024; i += 64) {  // lane-packed B operand table
    const int lane = i >> 5, idx = i & 31;
    const int ks = idx >> 3, t4 = (idx >> 1) & 3, v = idx & 1;
    const int ln = lane & 15, hi2 = (lane >> 4) * 2;
    bpack[i] = Acat[(4 * ks + hi2 + v) * 64 + 16 * t4 + ln];
  }
}

// ---------------------------------------------------------------------------
// Fused main kernel: one block per batch row.
//   patches -> encoder rows E -> F = E*A_cat via v_wmma_f32_16x16x4_f32
//   -> feats[784] in LDS -> 10x784 classifier + log_softmax
// ---------------------------------------------------------------------------
__global__ __launch_bounds__(256) void quanv_main_kernel(
    const float* __restrict__ x, const float* __restrict__ bpack,
    const float* __restrict__ W, const float* __restrict__ bias,
    float* __restrict__ out) {
  __shared__ float feats_sh[784];
  __shared__ float esh[8][16][20];  // per-wave encoder rows; 80B row stride
  __shared__ float fsh[8][16][68];  // per-wave F tiles; stride 68: halves on disjoint banks
  __shared__ float part_sh[10][16];
  __shared__ float logits_sh[10];

  const int b    = blockIdx.x;
  const int tid  = threadIdx.x;
  const int wave = tid >> 5;
  const int lane = tid & 31;
  const int ln   = lane & 15;
  const int hi   = lane >> 4;

  // Tile-invariant B operands: 16 v2f registers loaded once, reused all tiles.
  v2f breg[4][4];
#pragma unroll
  for (int ks = 0; ks < 4; ++ks)
#pragma unroll
    for (int t4 = 0; t4 < 4; ++t4) {
      const float* src = bpack + lane * 32 + ks * 8 + t4 * 2;
      breg[ks][t4].x = src[0];
      breg[ks][t4].y = src[1];
    }

  const float* xb = x + (long)b * 784;

  for (int tile = wave; tile < N_TILES; tile += 8) {  // wave-uniform loop
    const int p  = tile * 16 + ln;                    // patch row (dup per half)
    const int pc = p < N_PATCH ? p : N_PATCH - 1;     // clamp OOB rows
    const int pr = pc / 14, pcl = pc % 14;
    const float px0 = xb[(2 * pr + 0) * 28 + 2 * pcl + 0];
    const float px1 = xb[(2 * pr + 0) * 28 + 2 * pcl + 1];
    const float px2 = xb[(2 * pr + 1) * 28 + 2 * pcl + 0];
    const float px3 = xb[(2 * pr + 1) * 28 + 2 * pcl + 1];
    float s0, c0, s1, c1, s2, c2, s3, c3;
    __sincosf(0.5f * px0, &s0, &c0);
    __sincosf(0.5f * px1, &s1, &c1);
    __sincosf(0.5f * px2, &s2, &c2);
    __sincosf(0.5f * px3, &s3, &c3);
    float e[16];
#pragma unroll
    for (int m = 0; m < 16; ++m) {
      e[m] = ((m & 8) ? s0 : c0) * ((m & 4) ? s1 : c1) *
             ((m & 2) ? s2 : c2) * ((m & 1) ? s3 : c3);
    }
    if (hi == 0) {
      float4* ew = (float4*)&esh[wave][ln][0];
#pragma unroll
      for (int c = 0; c < 4; ++c)
        ew[c] = make_float4(e[4 * c + 0], e[4 * c + 1], e[4 * c + 2], e[4 * c + 3]);
    }

    // F(16x64) = E(16x16) * A_cat(16x64): 4 k-steps x 4 n-tiles.
    v8f Cacc[4];
#pragma unroll
    for (int t4 = 0; t4 < 4; ++t4) Cacc[t4] = (v8f){0, 0, 0, 0, 0, 0, 0, 0};
#pragma unroll
    for (int ks = 0; ks < 4; ++ks) {
      v2f a;  // constant-index selects: no dynamic register indexing
      a.x = hi ? e[4 * ks + 2] : e[4 * ks + 0];
      a.y = hi ? e[4 * ks + 3] : e[4 * ks + 1];
#pragma unroll
      for (int t4 = 0; t4 < 4; ++t4) {
        Cacc[t4] = __builtin_amdgcn_wmma_f32_16x16x4_f32(
            false, a, false, breg[ks][t4], (short)0, Cacc[t4], false, false);
      }
    }

    // Spill C tiles: lane holds (M = v + 8*hi, N = 16*t4 + ln).
#pragma unroll
    for (int t4 = 0; t4 < 4; ++t4)
#pragma unroll
      for (int v = 0; v < 8; ++v)
        fsh[wave][v + 8 * hi][16 * t4 + ln] = Cacc[t4][v];

    __asm__ volatile("" ::: "memory");  // LDS writes above before reads below

    // feats[r,w] = sum_k e_r[k] * F[r,16w+k]; 2 outputs per lane.
#pragma unroll
    for (int oo = 0; oo < 2; ++oo) {
      const int o = lane + 32 * oo;  // 0..63
      const int r = o >> 2, w = o & 3;
      const float4* ep = (const float4*)&esh[wave][r][0];
      const float4* fp = (const float4*)&fsh[wave][r][16 * w];
      float acc = 0.f;
#pragma unroll
      for (int c = 0; c < 4; ++c) {
        const float4 ev = ep[c];
        const float4 fv = fp[c];
        acc += ev.x * fv.x + ev.y * fv.y + ev.z * fv.z + ev.w * fv.w;
      }
      const int p2 = tile * 16 + r;
      if (p2 < N_PATCH) feats_sh[p2 * 4 + w] = acc;
    }
    __asm__ volatile("" ::: "memory");
  }

  __syncthreads();

  // logits[c] = bias[c] + sum_k feats[k] * W[c][k]
  if (tid < 160) {
    const int c = tid >> 4, g = tid & 15;
    float acc = 0.f;
    for (int k = g; k < 784; k += 16) acc += feats_sh[k] * W[c * 784 + k];
    part_sh[c][g] = acc;
  }
  __syncthreads();
  if (tid < 10) {
    float acc = bias[tid];
#pragma unroll
    for (int g = 0; g < 16; ++g) acc += part_sh[tid][g];
    logits_sh[tid] = acc;
  }
  __syncthreads();
  if (tid < 10) {
    float mx = -INFINITY;
#pragma unroll
    for (int c = 0; c < 10; ++c) mx = fmaxf(mx, logits_sh[c]);
    float se = 0.f;
#pragma unroll
    for (int c = 0; c < 10; ++c) se += expf(logits_sh[c] - mx);
    out[(long)b * 10 + tid] = logits_sh[tid] - mx - logf(se);
  }
}

// ---------------------------------------------------------------------------
extern "C" void kernel_launch(void* const* d_in, const int* in_sizes, int n_in,
                              void* d_out, int out_size, void* d_ws, size_t ws_size,
                              hipStream_t stream) {
  const float* x       = (const float*)d_in[0];  // (B,1,28,28)
  const float* vparams = (const float*)d_in[1];  // (3,8)
  const float* W       = (const float*)d_in[2];  // (10,784)
  const float* bvec    = (const float*)d_in[3];  // (10,)
  float* out = (float*)d_out;

  const int B = in_sizes[0] / 784;
  float* bpack = (float*)d_ws;  // 1024 floats = 4 KB scratch

  quanv_setup_kernel<<<1, 64, 0, stream>>>(vparams, bpack);
  quanv_main_kernel<<<B, 256, 0, stream>>>(x, bpack, W, bvec, out);
}